// CBP_60576218743166
// MI455X (gfx1250) — compile-verified
//
#include <hip/hip_runtime.h>
#include <hip/hip_bf16.h>
#include <math.h>

typedef __attribute__((ext_vector_type(2))) float v2f;
typedef __attribute__((ext_vector_type(8))) float v8f;

#define PROJ   8192
#define NCH    512
#define HWSZ   196      // 14*14, = 49 * 4 (K-steps of 4)
#define NTILE  1024     // (512/16)^2 gram tiles
#define BLK    256      // 8 wave32s

__global__ __launch_bounds__(BLK)
void cbp_gram_sketch_kernel(const float* __restrict__ x,
                            const float* __restrict__ s1,
                            const float* __restrict__ s2,
                            const int*   __restrict__ h1,
                            const int*   __restrict__ h2,
                            float* __restrict__ out)
{
    __shared__ float bins[PROJ];     // 32 KB sketch accumulator
    __shared__ float s1l[NCH], s2l[NCH];
    __shared__ int   h1l[NCH], h2l[NCH];
    __shared__ float red[8];

    const int b    = blockIdx.x;      // one workgroup per batch image
    const int tid  = threadIdx.x;
    const int lane = tid & 31;
    const int wave = tid >> 5;

    // ---- stage hash tables / signs, zero the bins ----
    for (int p = tid; p < PROJ; p += BLK) bins[p] = 0.0f;
    for (int c = tid; c < NCH; c += BLK) {
        s1l[c] = s1[c];  s2l[c] = s2[c];
        h1l[c] = h1[c];  h2l[c] = h2[c];
    }
    __syncthreads();

    const float* xb = x + (size_t)b * NCH * HWSZ;   // [512, 196] row-major

    const int half = lane >> 4;      // 0: lanes 0-15, 1: lanes 16-31
    const int lm16 = lane & 15;
    const int kb   = 2 * half;       // K sub-offset per ISA A/B f32 layout

    // ---- Gram tiles via V_WMMA_F32_16X16X4_F32, scatter into LDS bins ----
    for (int t = wave; t < NTILE; t += 8) {
        const int i = t >> 5;        // c-tile  (M)
        const int j = t & 31;        // c'-tile (N)

        const float* arow = xb + (size_t)(i * 16 + lm16) * HWSZ; // A: M = lane%16
        const float* brow = xb + (size_t)(j * 16 + lm16) * HWSZ; // B: N = lane%16

        v8f acc = {};
        #pragma unroll 7
        for (int kk = 0; kk < 49; ++kk) {
            const int k0 = kk * 4 + kb;
            v2f av = *(const v2f*)(arow + k0);   // K = k0, k0+1 (contiguous)
            v2f bv = *(const v2f*)(brow + k0);
            // 8 args: (neg_a, A, neg_b, B, c_mod, C, reuse_a, reuse_b)
            acc = __builtin_amdgcn_wmma_f32_16x16x4_f32(
                    false, av, false, bv, (short)0, acc, false, false);
        }

        // C/D layout: VGPR v, lane L -> M = v + 8*(L/16), N = L%16
        const int   cB  = j * 16 + lm16;   // c' (N) fixed per lane
        const float s2v = s2l[cB];
        const int   h2v = h2l[cB];
        #pragma unroll
        for (int v = 0; v < 8; ++v) {
            const int   cA  = i * 16 + v + 8 * half;          // c (M)
            const float val = acc[v] * s1l[cA] * s2v;
            const int   bkt = (h1l[cA] + h2v) & (PROJ - 1);   // mod 8192
            atomicAdd(&bins[bkt], val);                        // ds_add_f32
        }
    }
    __syncthreads();

    // ---- signed sqrt + L2 norm over the 8192 bins ----
    float yv[PROJ / BLK];
    float ss = 0.0f;
    #pragma unroll
    for (int q = 0; q < PROJ / BLK; ++q) {
        const float tv = bins[tid + q * BLK];
        const float sg = (tv > 0.0f) ? 1.0f : ((tv < 0.0f) ? -1.0f : 0.0f);
        const float y  = sg * sqrtf(fabsf(tv) + 1e-8f);
        yv[q] = y;
        ss   += y * y;
    }
    // wave32 reduction
    #pragma unroll
    for (int off = 16; off > 0; off >>= 1)
        ss += __shfl_down(ss, off, 32);
    if (lane == 0) red[wave] = ss;
    __syncthreads();
    if (tid == 0) {
        float tot = 0.0f;
        #pragma unroll
        for (int w = 0; w < 8; ++w) tot += red[w];
        red[0] = fmaxf(sqrtf(tot), 1e-12f);
    }
    __syncthreads();

    const float inv = 1.0f / red[0];
    float* ob = out + (size_t)b * PROJ;
    #pragma unroll
    for (int q = 0; q < PROJ / BLK; ++q)
        ob[tid + q * BLK] = yv[q] * inv;
}

extern "C" void kernel_launch(void* const* d_in, const int* in_sizes, int n_in,
                              void* d_out, int out_size, void* d_ws, size_t ws_size,
                              hipStream_t stream) {
    const float* x  = (const float*)d_in[0];   // [16, 512, 14, 14]
    const float* s1 = (const float*)d_in[1];   // [512]
    const float* s2 = (const float*)d_in[2];   // [512]
    const int*   h1 = (const int*)  d_in[3];   // [512]
    const int*   h2 = (const int*)  d_in[4];   // [512]
    float*       y  = (float*)d_out;           // [16, 8192]
    (void)in_sizes; (void)n_in; (void)out_size; (void)d_ws; (void)ws_size;

    cbp_gram_sketch_kernel<<<16, BLK, 0, stream>>>(x, s1, s2, h1, h2, y);
}